// Conv2D_25134148616970
// MI455X (gfx1250) — compile-verified
//
#include <hip/hip_runtime.h>

typedef float v2f  __attribute__((ext_vector_type(2)));
typedef float v8f  __attribute__((ext_vector_type(8)));
typedef unsigned int v4u __attribute__((ext_vector_type(4)));
typedef int   v4i  __attribute__((ext_vector_type(4)));
typedef int   v8i  __attribute__((ext_vector_type(8)));

#define IN_H   4096
#define IN_W   8192
#define OUT_H  4094
#define OUT_W  8190

#define BR     64            // output rows per block
#define BC     128           // output cols per block
#define TILE_H 66            // BR + 2 halo
#define TILE_W 132           // BC + 2 halo + K-padding room; 132 % 64 = 4-bank row skew

__global__ __launch_bounds__(256)
void conv3x3_tdm_wmma(const float* __restrict__ X,
                      const float* __restrict__ Wk,
                      const float* __restrict__ bias,
                      float* __restrict__ Y) {
    // wpad FIRST so its 120 floats sit at LDS offset 0: all banded-B loads then
    // use immediate ds offsets (dr*40+4c <= 97 < 256) off one base register.
    __shared__ float wpad[3 * 40];            // banded-weight lookup, zero padded
    __shared__ float tile[TILE_H * TILE_W];   // 34848 B, written by TDM (16B-aligned base)

    const int row0 = blockIdx.y * BR;
    const int col0 = blockIdx.x * BC;

    const int tid  = threadIdx.x;
    const int lane = tid & 31;
    const int wave = tid >> 5;
    const int half = lane >> 4;      // 0: lanes 0-15, 1: lanes 16-31
    const int lo16 = lane & 15;

    // ---- padded weight table: wpad[dr][16+d] = W[dr][d] for d in 0..2, else 0
    if (tid < 120) {
        const int dr  = tid / 40;
        const int idx = tid - dr * 40;
        float v = 0.0f;
        if (idx >= 16 && idx <= 18) v = Wk[dr * 3 + (idx - 16)];
        wpad[tid] = v;
    }

    // ---------------- TDM: stage (row0..+65, col0..+131) -> LDS ---------------
    // OOB-past-tensor reads return zero => free right/bottom edge handling.
    if (wave == 0) {
        unsigned long long gaddr =
            (unsigned long long)(const void*)(X + (size_t)row0 * IN_W + col0);
        unsigned int lds_base = (unsigned int)(unsigned long long)(const void*)&tile[0];

        v4u g0;
        g0.x = 1u;                                               // count=1
        g0.y = lds_base;
        g0.z = (unsigned int)(gaddr & 0xFFFFFFFFull);
        g0.w = (unsigned int)((gaddr >> 32) & 0x1FFFFFFull) | (2u << 30); // type=2

        unsigned int tdim0 = (unsigned int)(IN_W - col0);        // remaining width
        unsigned int tdim1 = (unsigned int)(IN_H - row0);        // remaining height

        v8i g1;
        g1[0] = (int)(2u << 16);                                   // data_size=4B
        g1[1] = (int)((tdim0 & 0xFFFFu) << 16);                    // tensor_dim0
        g1[2] = (int)((tdim0 >> 16) | ((tdim1 & 0xFFFFu) << 16));  // tensor_dim1
        g1[3] = (int)((tdim1 >> 16) | ((unsigned)TILE_W << 16));   // tile_dim0
        g1[4] = (int)TILE_H;                                       // tile_dim1
        g1[5] = (int)IN_W;                                         // dim0 stride
        g1[6] = 0;
        g1[7] = 0;

        v4i z4 = {0, 0, 0, 0};
#if defined(__clang_major__) && __clang_major__ >= 23
        v8i z8 = {0, 0, 0, 0, 0, 0, 0, 0};
        __builtin_amdgcn_tensor_load_to_lds(g0, g1, z4, z4, z8, 0);
#else
        __builtin_amdgcn_tensor_load_to_lds(g0, g1, z4, z4, 0);
#endif
        __builtin_amdgcn_s_wait_tensorcnt(0);
    }

    const float bval = bias[0];

    __syncthreads();

    // ---- B matrices from LDS table (B[dr][c] layout: K=2*half+{0,1}, N=lo16) --
    // element index = (dr*40) + 16 + (4c + 2*half) - n ; variable part in [1,35]
    const float* bbase = &wpad[16 + 2 * half - lo16];
    v2f B[3][5];
#pragma unroll
    for (int dr = 0; dr < 3; ++dr) {
#pragma unroll
        for (int c = 0; c < 5; ++c) {
            const float* p = bbase + dr * 40 + 4 * c;   // immediate ds offsets
            B[dr][c].x = p[0];
            B[dr][c].y = p[1];
        }
    }

    // ------------- per-wave 32x32 outputs: 4 tiles x 15 WMMAs each ------------
    const int wr = wave >> 2;     // 0..1  (row group of 32)
    const int wc = wave & 3;      // 0..3  (col group of 32)

    const float* abase = &tile[(32 * wr + lo16) * TILE_W + 32 * wc + 2 * half];

    v8f acc00 = {};   // rows +0,  cols +0
    v8f acc01 = {};   // rows +0,  cols +16
    v8f acc10 = {};   // rows +16, cols +0
    v8f acc11 = {};   // rows +16, cols +16
#pragma unroll
    for (int dr = 0; dr < 3; ++dr) {
#pragma unroll
        for (int c = 0; c < 5; ++c) {
            const float* p0 = abase + dr * TILE_W + 4 * c;        // 8B-aligned
            const float* p1 = p0 + 16 * TILE_W;
            v2f a00 = *(const v2f*)(p0);
            v2f a01 = *(const v2f*)(p0 + 16);
            v2f a10 = *(const v2f*)(p1);
            v2f a11 = *(const v2f*)(p1 + 16);
            acc00 = __builtin_amdgcn_wmma_f32_16x16x4_f32(
                        false, a00, false, B[dr][c], (short)0, acc00, false, false);
            acc01 = __builtin_amdgcn_wmma_f32_16x16x4_f32(
                        false, a01, false, B[dr][c], (short)0, acc01, false, false);
            acc10 = __builtin_amdgcn_wmma_f32_16x16x4_f32(
                        false, a10, false, B[dr][c], (short)0, acc10, false, false);
            acc11 = __builtin_amdgcn_wmma_f32_16x16x4_f32(
                        false, a11, false, B[dr][c], (short)0, acc11, false, false);
        }
    }

    // ---------------- bias + store (C/D layout: M = j + 8*half) ---------------
    const int orow0 = row0 + 32 * wr;
    const int ocol0 = col0 + 32 * wc + lo16;
    const bool interior = (row0 + BR <= OUT_H) && (col0 + BC <= OUT_W);

    if (interior) {
#pragma unroll
        for (int j = 0; j < 8; ++j) {
            const size_t rA = (size_t)(orow0 + j + 8 * half) * OUT_W;
            const size_t rB = (size_t)(orow0 + 16 + j + 8 * half) * OUT_W;
            Y[rA + ocol0]      = acc00[j] + bval;
            Y[rA + ocol0 + 16] = acc01[j] + bval;
            Y[rB + ocol0]      = acc10[j] + bval;
            Y[rB + ocol0 + 16] = acc11[j] + bval;
        }
    } else {
#pragma unroll
        for (int j = 0; j < 8; ++j) {
            const int rA = orow0 + j + 8 * half;
            const int rB = rA + 16;
            const bool cok0 = (ocol0 < OUT_W);
            const bool cok1 = (ocol0 + 16 < OUT_W);
            if (rA < OUT_H) {
                const size_t r = (size_t)rA * OUT_W;
                if (cok0) Y[r + ocol0]      = acc00[j] + bval;
                if (cok1) Y[r + ocol0 + 16] = acc01[j] + bval;
            }
            if (rB < OUT_H) {
                const size_t r = (size_t)rB * OUT_W;
                if (cok0) Y[r + ocol0]      = acc10[j] + bval;
                if (cok1) Y[r + ocol0 + 16] = acc11[j] + bval;
            }
        }
    }
}

extern "C" void kernel_launch(void* const* d_in, const int* in_sizes, int n_in,
                              void* d_out, int out_size, void* d_ws, size_t ws_size,
                              hipStream_t stream) {
    const float* X  = (const float*)d_in[0];
    const float* Wk = (const float*)d_in[1];
    const float* b  = (const float*)d_in[2];
    float* Y        = (float*)d_out;

    dim3 grid((OUT_W + BC - 1) / BC, (OUT_H + BR - 1) / BR);  // 64 x 64
    conv3x3_tdm_wmma<<<grid, dim3(256), 0, stream>>>(X, Wk, b, Y);
}